// TAGConvModel_42863773614471
// MI455X (gfx1250) — compile-verified
//
#include <hip/hip_runtime.h>
#include <hip/hip_bf16.h>

// ---------------------------------------------------------------------------
// TAGConv (K=2) x2 + mean-pool + linear classifier for MI455X (gfx1250).
// Dense GEMMs (N x 384 @ 384 x 128, twice) dominate FLOPs -> V_WMMA_F32_16X16X4_F32
// (full fp32 precision, wave32). Each wave owns a 64x32 output macro-tile
// (4x2 WMMA tiles): 8 independent wmma per K=4 step against 8 load instrs.
// SpMM hops use global f32 atomics; working set (~150MB) is L2-resident.
// ---------------------------------------------------------------------------

#define N_NODES 50000
#define N_EDGES 600000
#define NB      64
#define DIN     128
#define DH      128
#define PF      32
#define NC      2

typedef __attribute__((ext_vector_type(2))) float v2f;
typedef __attribute__((ext_vector_type(8))) float v8f;

__global__ void zero_f32(float* __restrict__ p, long long n) {
    long long i = (long long)blockIdx.x * blockDim.x + threadIdx.x;
    if (i < n) p[i] = 0.0f;
}

__global__ void deg_scatter(const int* __restrict__ dst, float* __restrict__ deg) {
    int e = blockIdx.x * blockDim.x + threadIdx.x;
    if (e < N_EDGES) atomicAdd(&deg[dst[e]], 1.0f);
}

__global__ void make_norm(const float* __restrict__ deg, float* __restrict__ norm) {
    int i = blockIdx.x * blockDim.x + threadIdx.x;
    if (i < N_NODES) norm[i] = rsqrtf(fmaxf(deg[i], 1.0f));
}

// out[dst] += x[src] * norm[src]; one thread handles 4 contiguous features.
__global__ void spmm_scatter(const float* __restrict__ x, const int* __restrict__ src,
                             const int* __restrict__ dst, const float* __restrict__ norm,
                             float* __restrict__ out) {
    long long idx = (long long)blockIdx.x * blockDim.x + threadIdx.x;
    int e = (int)(idx >> 5);
    if (e >= N_EDGES) return;
    int f = ((int)idx & 31) << 2;
    int s = src[e], d = dst[e];
    float ns = norm[s];
    const float4 xv = *(const float4*)(x + (size_t)s * DIN + f);
    float* op = out + (size_t)d * DIN + f;
    atomicAdd(op + 0, xv.x * ns);
    atomicAdd(op + 1, xv.y * ns);
    atomicAdd(op + 2, xv.z * ns);
    atomicAdd(op + 3, xv.w * ns);
}

// x[n][:] *= norm[n]
__global__ void scale_rows(float* __restrict__ x, const float* __restrict__ norm) {
    long long idx = (long long)blockIdx.x * blockDim.x + threadIdx.x;
    if (idx < (long long)N_NODES * DIN) {
        int row = (int)(idx >> 7);
        x[idx] *= norm[row];
    }
}

// out = relu?([X0|X1|X2] @ W + b). W is [3*128, 128] row-major.
// Block = 128 threads = 4 waves; wave w -> cols [32w, 32w+32), block -> 64 rows.
// Each wave holds a 4(M) x 2(N) grid of 16x16 f32 accumulators and runs
// v_wmma_f32_16x16x4_f32 over K=384 in steps of 4:
//   per step: 4 x b64 A loads + 2 x (2 b32) B loads -> 8 independent WMMAs.
// Fragment layouts (f32):
//   A 16x4 : lanes 0-15 = rows; VGPR0 = {K0 | K2}, VGPR1 = {K1 | K3}
//   B 4x16 : lanes = cols;      VGPR0 = {K0 | K2}, VGPR1 = {K1 | K3}
//   C 16x16: VGPR r = row r (lanes 0-15) / row r+8 (lanes 16-31)
// OOB rows (last block) are clamped at load (EXEC must stay all-ones for WMMA)
// and masked at store.
__global__ __launch_bounds__(128)
void tag_gemm(const float* __restrict__ X0, const float* __restrict__ X1,
              const float* __restrict__ X2, const float* __restrict__ W,
              const float* __restrict__ bias, float* __restrict__ out, int do_relu) {
    const int lane = threadIdx.x & 31;
    const int wave = threadIdx.x >> 5;       // 0..3 -> 32-col group
    const int m0   = blockIdx.x * 64;        // 4 row tiles of 16
    const int n0   = wave * 32;              // 2 col tiles of 16
    const int lrow = lane & 15;
    const int khi  = (lane >> 4) << 1;       // 0 for lanes 0-15, 2 for 16-31

    const float* srcs[3] = {X0, X1, X2};

    v8f acc[4][2];
#pragma unroll
    for (int mt = 0; mt < 4; ++mt)
#pragma unroll
        for (int nt = 0; nt < 2; ++nt)
            acc[mt][nt] = (v8f){0.f, 0.f, 0.f, 0.f, 0.f, 0.f, 0.f, 0.f};

    // Per-tile A row indices, clamped for the partial last block.
    int arow[4];
#pragma unroll
    for (int mt = 0; mt < 4; ++mt) {
        int r = m0 + mt * 16 + lrow;
        arow[mt] = (r < N_NODES) ? r : (N_NODES - 1);
    }

    for (int s = 0; s < 3; ++s) {
        const float* Xs  = srcs[s];
        const float* Ap0 = Xs + (size_t)arow[0] * DIN + khi;
        const float* Ap1 = Xs + (size_t)arow[1] * DIN + khi;
        const float* Ap2 = Xs + (size_t)arow[2] * DIN + khi;
        const float* Ap3 = Xs + (size_t)arow[3] * DIN + khi;
        const float* Wp  = W + (size_t)(s * DIN + khi) * DH + n0 + lrow;
#pragma unroll 2
        for (int k = 0; k < DIN; k += 4) {
            v2f a0, a1, a2, a3, b0, b1;
            a0.x = Ap0[k]; a0.y = Ap0[k + 1];
            a1.x = Ap1[k]; a1.y = Ap1[k + 1];
            a2.x = Ap2[k]; a2.y = Ap2[k + 1];
            a3.x = Ap3[k]; a3.y = Ap3[k + 1];
            const size_t wo = (size_t)k * DH;
            b0.x = Wp[wo];      b0.y = Wp[wo + DH];        // cols n0..n0+15
            b1.x = Wp[wo + 16]; b1.y = Wp[wo + DH + 16];   // cols n0+16..n0+31
            acc[0][0] = __builtin_amdgcn_wmma_f32_16x16x4_f32(false, a0, false, b0, (short)0, acc[0][0], false, false);
            acc[0][1] = __builtin_amdgcn_wmma_f32_16x16x4_f32(false, a0, false, b1, (short)0, acc[0][1], false, false);
            acc[1][0] = __builtin_amdgcn_wmma_f32_16x16x4_f32(false, a1, false, b0, (short)0, acc[1][0], false, false);
            acc[1][1] = __builtin_amdgcn_wmma_f32_16x16x4_f32(false, a1, false, b1, (short)0, acc[1][1], false, false);
            acc[2][0] = __builtin_amdgcn_wmma_f32_16x16x4_f32(false, a2, false, b0, (short)0, acc[2][0], false, false);
            acc[2][1] = __builtin_amdgcn_wmma_f32_16x16x4_f32(false, a2, false, b1, (short)0, acc[2][1], false, false);
            acc[3][0] = __builtin_amdgcn_wmma_f32_16x16x4_f32(false, a3, false, b0, (short)0, acc[3][0], false, false);
            acc[3][1] = __builtin_amdgcn_wmma_f32_16x16x4_f32(false, a3, false, b1, (short)0, acc[3][1], false, false);
        }
    }

    // Epilogue: bias + optional ReLU, guarded stores for the partial block.
    const int coff = (lane >> 4) << 3;   // +0 (lanes 0-15) or +8 (lanes 16-31)
#pragma unroll
    for (int nt = 0; nt < 2; ++nt) {
        const int col = n0 + nt * 16 + lrow;
        const float bv = bias[col];
#pragma unroll
        for (int mt = 0; mt < 4; ++mt) {
            const int mbase = m0 + mt * 16 + coff;
#pragma unroll
            for (int r = 0; r < 8; ++r) {
                const int row = mbase + r;
                if (row < N_NODES) {
                    float v = acc[mt][nt][r] + bv;
                    if (do_relu) v = fmaxf(v, 0.0f);
                    out[(size_t)row * DH + col] = v;
                }
            }
        }
    }
}

// sums[gid[n]] += x[n]; cnt[gid[n]] += 1
__global__ void pool_scatter(const float* __restrict__ x, const int* __restrict__ gid,
                             float* __restrict__ sums, float* __restrict__ cnt) {
    long long idx = (long long)blockIdx.x * blockDim.x + threadIdx.x;
    int n = (int)(idx >> 5);
    if (n >= N_NODES) return;
    int f = ((int)idx & 31) << 2;
    int g = gid[n];
    const float4 xv = *(const float4*)(x + (size_t)n * DH + f);
    float* sp = sums + (size_t)g * DH + f;
    atomicAdd(sp + 0, xv.x);
    atomicAdd(sp + 1, xv.y);
    atomicAdd(sp + 2, xv.z);
    atomicAdd(sp + 3, xv.w);
    if (f == 0) atomicAdd(&cnt[g], 1.0f);
}

// out[b][c] = bc[c] + mean_row(b) . Wc[0:128, c] + perm[b] . Wc[128:160, c]
__global__ void classify(const float* __restrict__ sums, const float* __restrict__ cnt,
                         const float* __restrict__ perm, const float* __restrict__ Wc,
                         const float* __restrict__ bc, float* __restrict__ out) {
    int t = threadIdx.x;
    if (t >= NB * NC) return;
    int b = t >> 1, c = t & 1;
    float inv = 1.0f / fmaxf(cnt[b], 1.0f);
    float acc = bc[c];
    for (int j = 0; j < DH; ++j) acc += sums[b * DH + j] * inv * Wc[j * NC + c];
    for (int j = 0; j < PF; ++j) acc += perm[b * PF + j] * Wc[(DH + j) * NC + c];
    out[t] = acc;
}

extern "C" void kernel_launch(void* const* d_in, const int* in_sizes, int n_in,
                              void* d_out, int out_size, void* d_ws, size_t ws_size,
                              hipStream_t stream) {
    const float* h   = (const float*)d_in[0];
    const int*   src = (const int*)d_in[1];
    const int*   dst = (const int*)d_in[2];
    const int*   gid = (const int*)d_in[3];
    const float* pf  = (const float*)d_in[4];
    const float* W1  = (const float*)d_in[5];
    const float* b1  = (const float*)d_in[6];
    const float* W2  = (const float*)d_in[7];
    const float* b2  = (const float*)d_in[8];
    const float* Wc  = (const float*)d_in[9];
    const float* bc  = (const float*)d_in[10];
    float* out = (float*)d_out;

    // Workspace carve-up (fp32). 2*50000 floats prefix keeps 16B alignment.
    float* ws   = (float*)d_ws;
    float* deg  = ws;  ws += N_NODES;
    float* norm = ws;  ws += N_NODES;
    const size_t NF = (size_t)N_NODES * DIN;
    float* a1   = ws;  ws += NF;   // hop-1 buffer (reused across layers)
    float* a2   = ws;  ws += NF;   // hop-2 buffer (reused across layers)
    float* x1   = ws;  ws += NF;   // TAGConv-1 output
    float* x2   = ws;  ws += NF;   // TAGConv-2 output
    float* sums = ws;  ws += NB * DH;
    float* cnt  = ws;

    const int T = 256;
    const int gN    = (N_NODES + T - 1) / T;
    const int gE    = (N_EDGES + T - 1) / T;
    const int gE32  = (int)(((long long)N_EDGES * 32 + T - 1) / T);
    const int gNF   = (int)((NF + T - 1) / T);
    const int gN32  = (int)(((long long)N_NODES * 32 + T - 1) / T);
    const int gGemm = (N_NODES + 63) / 64;   // 782 blocks of 128 threads

    // 1) degree + norm
    zero_f32<<<gN, T, 0, stream>>>(deg, N_NODES);
    deg_scatter<<<gE, T, 0, stream>>>(dst, deg);
    make_norm<<<gN, T, 0, stream>>>(deg, norm);

    // 2) TAGConv layer 1: a1 = A_n h, a2 = A_n a1
    zero_f32<<<2 * gNF, T, 0, stream>>>(a1, 2 * (long long)NF);   // a1,a2 contiguous
    spmm_scatter<<<gE32, T, 0, stream>>>(h, src, dst, norm, a1);
    scale_rows<<<gNF, T, 0, stream>>>(a1, norm);
    spmm_scatter<<<gE32, T, 0, stream>>>(a1, src, dst, norm, a2);
    scale_rows<<<gNF, T, 0, stream>>>(a2, norm);
    tag_gemm<<<gGemm, 128, 0, stream>>>(h, a1, a2, W1, b1, x1, 1);

    // 3) TAGConv layer 2
    zero_f32<<<2 * gNF, T, 0, stream>>>(a1, 2 * (long long)NF);
    spmm_scatter<<<gE32, T, 0, stream>>>(x1, src, dst, norm, a1);
    scale_rows<<<gNF, T, 0, stream>>>(a1, norm);
    spmm_scatter<<<gE32, T, 0, stream>>>(a1, src, dst, norm, a2);
    scale_rows<<<gNF, T, 0, stream>>>(a2, norm);
    tag_gemm<<<gGemm, 128, 0, stream>>>(x1, a1, a2, W2, b2, x2, 0);

    // 4) mean-pool + classifier
    zero_f32<<<(NB * DH + NB + T - 1) / T, T, 0, stream>>>(sums, NB * DH + NB);
    pool_scatter<<<gN32, T, 0, stream>>>(x2, gid, sums, cnt);
    classify<<<1, 128, 0, stream>>>(sums, cnt, pf, Wc, bc, out);

    (void)in_sizes; (void)n_in; (void)out_size; (void)ws_size;
}